// InstantNGPModel_17514876634260
// MI455X (gfx1250) — compile-verified
//
#include <hip/hip_runtime.h>
#include <stdint.h>

// ---------------------------------------------------------------------------
// InstantNGP multi-level hash-grid encode for gfx1250 (MI455X).
//
// Key facts exploited:
//  * Reference applies HASH_MOD = 4096 to ALL levels -> only the first
//    4096 entries (32 KB) of each level's table are ever read.
//  * 4096 is a power of two -> modulo becomes '& 4095'.
//  * RESOLUTIONS[i] == 16 << i exactly.
//  * Output layout: out[n*32 + level*2 + f], float32.
//
// Strategy: blockIdx.y = level. Each block DMAs the 32 KB hot table slice
// into LDS via the Tensor Data Mover (tensor_load_to_lds + s_wait_tensorcnt),
// then each thread processes 8 points: hash 8 corners, ds_load_b64 gather
// from LDS, trilinear blend (reference's reversed-weight convention),
// global_store_b64 the 2-feature result.
// ---------------------------------------------------------------------------

#ifndef __has_builtin
#define __has_builtin(x) 0
#endif

#define NGP_HAVE_TDM (__has_builtin(__builtin_amdgcn_tensor_load_to_lds) && \
                      __has_builtin(__builtin_amdgcn_s_wait_tensorcnt))

typedef uint32_t u32;
typedef __attribute__((ext_vector_type(4))) u32 v4u;
typedef __attribute__((ext_vector_type(8))) int v8i;
typedef __attribute__((ext_vector_type(4))) int v4i;

struct TablePtrs { const float* t[16]; };

#define STAGE_FLOATS 8192        // 4096 entries * 2 feats
#define HASH_MASK    4095u
#define PRIME_Y      2654435761u
#define PRIME_Z      805459861u

static __device__ __forceinline__ int clampi(int v, int lo, int hi) {
    return v < lo ? lo : (v > hi ? hi : v);
}

__global__ __launch_bounds__(256)
void ngp_encode_kernel(const float* __restrict__ pos, TablePtrs tabs,
                       float* __restrict__ out, int n_points)
{
    const int level = blockIdx.y;
    const int R     = 16 << level;          // exact: ceil(16 * 2^i)
    const int Rm    = R - 1;

    __shared__ float tab[STAGE_FLOATS];     // 32 KB hot table slice
    const float* __restrict__ gt = tabs.t[level];

#if NGP_HAVE_TDM
    // One wave issues the TDM descriptor; EXEC is ignored by TENSOR ops,
    // so we branch so that only wave 0 reaches the instruction.
    if ((threadIdx.x >> 5) == 0) {
        const uint64_t ga = (uint64_t)(uintptr_t)gt;       // global tile start
        const u32 lds_base = (u32)(uintptr_t)(&tab[0]);    // addr[31:0] = LDS byte offset

        v4u g0;
        g0[0] = 1u;                                        // count=1, user descriptor
        g0[1] = lds_base;                                  // lds_addr
        g0[2] = (u32)ga;                                   // global_addr[31:0]
        g0[3] = (u32)((ga >> 32) & 0x01FFFFFFu)            // global_addr[56:32]
              | (2u << 30);                                // type = 2 ("image")

        v8i g1;
        g1[0] = (int)(2u << 16);        // workgroup_mask=0 (not in cluster), data_size=4B
        g1[1] = (int)(0x2000u << 16);   // tensor_dim0 = 8192 (low 16 bits at [63:48])
        g1[2] = (int)(1u << 16);        // tensor_dim0 hi=0; tensor_dim1 = 1
        g1[3] = (int)(0x2000u << 16);   // tensor_dim1 hi=0; tile_dim0 = 8192
        g1[4] = 1;                      // tile_dim1 = 1; tile_dim2 = 0
        g1[5] = 0x2000;                 // tensor_dim0_stride = 8192 (low 32)
        g1[6] = 0;                      // stride hi / dim1_stride lo
        g1[7] = 0;

        v4i zero4 = {0, 0, 0, 0};       // groups 2/3 unused for 2D tile
#if __clang_major__ >= 23
        v8i zero8 = {0, 0, 0, 0, 0, 0, 0, 0};
        __builtin_amdgcn_tensor_load_to_lds(g0, g1, zero4, zero4, zero8, 0);
#else
        __builtin_amdgcn_tensor_load_to_lds(g0, g1, zero4, zero4, 0);
#endif
        __builtin_amdgcn_s_wait_tensorcnt(0);
        asm volatile("" ::: "memory");  // LDS now holds the staged table
    }
#else
    // Fallback: cooperative vectorized copy (b128 loads + ds_store_b128).
    {
        const float4* __restrict__ src = reinterpret_cast<const float4*>(gt);
        float4* dst = reinterpret_cast<float4*>(tab);
        for (int i = threadIdx.x; i < STAGE_FLOATS / 4; i += blockDim.x)
            dst[i] = src[i];
    }
#endif
    __syncthreads();

    const float rs = 0.5f * (float)Rm;
    const int stride = gridDim.x * blockDim.x;

    for (int n = blockIdx.x * blockDim.x + threadIdx.x; n < n_points; n += stride) {
        // Prefetch next iteration's position data into cache (global_prefetch_b8).
        if (n + stride < n_points)
            __builtin_prefetch(&pos[3 * (size_t)(n + stride)], 0, 3);

        const float px = pos[3 * (size_t)n + 0];
        const float py = pos[3 * (size_t)n + 1];
        const float pz = pos[3 * (size_t)n + 2];

        const float sx = (px + 1.0f) * rs;
        const float sy = (py + 1.0f) * rs;
        const float sz = (pz + 1.0f) * rs;
        const float fx = floorf(sx), fy = floorf(sy), fz = floorf(sz);
        const float wx = sx - fx, wy = sy - fy, wz = sz - fz;

        const int gx = (int)fx, gy = (int)fy, gz = (int)fz;
        const u32 x0 = (u32)clampi(gx,     0, Rm);
        const u32 x1 = (u32)clampi(gx + 1, 0, Rm);
        const u32 y0 = (u32)clampi(gy,     0, Rm);
        const u32 y1 = (u32)clampi(gy + 1, 0, Rm);
        const u32 z0 = (u32)clampi(gz,     0, Rm);
        const u32 z1 = (u32)clampi(gz + 1, 0, Rm);

        // h = (x*1) ^ (y*P1) ^ (z*P2), idx = h & 4095
        const u32 hx0 = x0,            hx1 = x1;
        const u32 hy0 = y0 * PRIME_Y,  hy1 = y1 * PRIME_Y;
        const u32 hz0 = z0 * PRIME_Z,  hz1 = z1 * PRIME_Z;

        // Reference uses reversed weights: offset comp dx pairs with w[2]=wz,
        // dy with w[1]=wy, dz with w[0]=wx.
        const float ax0 = 1.0f - wz, ax1 = wz;   // selected by dx
        const float ay0 = 1.0f - wy, ay1 = wy;   // selected by dy
        const float az0 = 1.0f - wx, az1 = wx;   // selected by dz

        float o0 = 0.0f, o1 = 0.0f;
        #pragma unroll
        for (int dx = 0; dx < 2; ++dx) {
            const u32   hx = dx ? hx1 : hx0;
            const float wa = dx ? ax1 : ax0;
            #pragma unroll
            for (int dy = 0; dy < 2; ++dy) {
                const u32   hxy = hx ^ (dy ? hy1 : hy0);
                const float wab = wa * (dy ? ay1 : ay0);
                #pragma unroll
                for (int dz = 0; dz < 2; ++dz) {
                    const u32   idx = (hxy ^ (dz ? hz1 : hz0)) & HASH_MASK;
                    const float cw  = wab * (dz ? az1 : az0);
                    // 8-byte LDS gather (ds_load_b64): both feats of this corner.
                    const float2 f = *reinterpret_cast<const float2*>(&tab[idx * 2]);
                    o0 = fmaf(cw, f.x, o0);
                    o1 = fmaf(cw, f.y, o1);
                }
            }
        }

        // 8-byte coalescable store: out[n][level*2 .. level*2+1]
        float2 r; r.x = o0; r.y = o1;
        *reinterpret_cast<float2*>(&out[(size_t)n * 32 + (size_t)level * 2]) = r;
    }
}

extern "C" void kernel_launch(void* const* d_in, const int* in_sizes, int n_in,
                              void* d_out, int out_size, void* d_ws, size_t ws_size,
                              hipStream_t stream) {
    (void)d_ws; (void)ws_size; (void)out_size; (void)n_in;

    const float* pos = (const float*)d_in[0];
    TablePtrs tabs;
    for (int i = 0; i < 16; ++i)
        tabs.t[i] = (const float*)d_in[1 + i];
    float* out = (float*)d_out;

    const int n_points = in_sizes[0] / 3;           // 1,048,576
    const int threads  = 256;
    const int ppt      = 8;                          // points per thread
    const int gx       = (n_points + threads * ppt - 1) / (threads * ppt);  // 512

    dim3 grid((unsigned)gx, 16, 1);                 // y = level
    ngp_encode_kernel<<<grid, dim3(threads, 1, 1), 0, stream>>>(pos, tabs, out, n_points);
}